// RingEdgeEncoderOld_46660524703963
// MI455X (gfx1250) — compile-verified
//
#include <hip/hip_runtime.h>

// Problem constants (fixed by the reference: B=32, N=256, EMB=32, E=32768).
#define BB    32
#define NN    256
#define EMB   32
#define CELLS (BB * NN * NN)        // 2,097,152 (b,i,j) cells
#define MASK_WORDS (CELLS / 32)     // 65,536 u32 words = 256 KB bitmask in d_ws

typedef float        f4 __attribute__((ext_vector_type(4)));
typedef unsigned int u4 __attribute__((ext_vector_type(4)));

// ---------------------------------------------------------------------------
// HOT KERNEL (placed first so the disasm snippet shows it).
// One float4 per thread; 8 threads per EMB=32 cell. A 256-thread block covers
// exactly 32 cells == one mask word, so mask[blockIdx.x] is a uniform address
// -> scalar load (s_load_b32, KMcnt), zero per-lane VMEM for the mask.
// 512 MB total traffic -> ~22 us floor at 23.3 TB/s. Input+output (512 MB)
// exceed the 192 MB L2, so both the load and the store use non-temporal
// hints to keep the stream from evicting the hot mask/weight lines.
// ---------------------------------------------------------------------------
__global__ void __launch_bounds__(256)
ring_stream_add(const float* __restrict__ in,
                const float* __restrict__ w,       // emb_weight [2][EMB]
                const unsigned int* __restrict__ mask,
                float* __restrict__ out) {
    unsigned int t    = blockIdx.x * 256u + threadIdx.x;  // global float4 index
    unsigned int lane = threadIdx.x & 7u;                 // float4 within cell
    unsigned int cib  = threadIdx.x >> 3;                 // cell within block

    // Uniform address: compiler emits a scalar load for the whole block.
    unsigned int word = mask[blockIdx.x];
    float scale = (float)((word >> cib) & 1u);

    const f4* __restrict__ in4 = (const f4*)in;

    // Lane-0-per-wave speculative prefetch ~128 KB ahead of the stream
    // (gfx1250 global_prefetch_b8; invalid-translation prefetches are dropped).
    if ((threadIdx.x & 31u) == 0u) {
        __builtin_prefetch((const void*)(in4 + t + 8192u), 0, 0);
    }

    f4 v = __builtin_nontemporal_load(in4 + t);           // global_load_b128 NT

    // Row 1 of the embedding (row 0 is forced to the zero vector).
    const f4* __restrict__ w1 = (const f4*)(w + EMB);
    f4 wv = w1[lane];                                     // 128 B, cache-hot

    v += wv * scale;                                      // exact pass-through when scale==0

    __builtin_nontemporal_store(v, ((f4*)out) + t);       // global_store_b128 NT
}

// ---------------------------------------------------------------------------
// Kernel 1: zero the adjacency bitmask (runs every call: d_ws is poisoned
// once and never re-poisoned; we own re-initialization; no static state).
// uint4 stores: 16,384 threads, 64 blocks.
// ---------------------------------------------------------------------------
__global__ void __launch_bounds__(256)
ring_zero_mask(unsigned int* __restrict__ mask) {
    unsigned int i = blockIdx.x * 256u + threadIdx.x;     // uint4 index
    u4 z = (u4)0u;
    ((u4*)mask)[i] = z;
}

// ---------------------------------------------------------------------------
// Kernel 2: scatter E edges into the bitmask.
// adj[g, ls, ld] = 1  ->  bit (g*N*N + ls*N + ld) of mask (global_atomic_or_b32).
// ---------------------------------------------------------------------------
__global__ void __launch_bounds__(256)
ring_scatter_edges(const int* __restrict__ ring,
                   const int* __restrict__ batch,
                   unsigned int* __restrict__ mask,
                   int E) {
    int e = blockIdx.x * 256 + threadIdx.x;
    if (e >= E) return;
    int src = ring[e];          // ring_index row 0
    int dst = ring[E + e];      // ring_index row 1
    int g   = batch[src];       // graph id of the edge
    int ls  = src - g * NN;     // local indices (equal-size graphs)
    int ld  = dst - g * NN;
    unsigned int lin = ((unsigned int)(g * NN + ls)) * (unsigned int)NN
                     + (unsigned int)ld;                  // g*N*N + ls*N + ld
    atomicOr(&mask[lin >> 5], 1u << (lin & 31u));
}

// ---------------------------------------------------------------------------
// Host launcher. Inputs (setup_inputs order):
//   d_in[0] edge_dense f32 [B,N,N,EMB]
//   d_in[1] emb_weight f32 [2,EMB]
//   d_in[2] ring_index i32 [2,E]
//   d_in[3] batch_vec  i32 [B*N]
// d_out: f32 [B,N,N,EMB].  d_ws: 256 KB bitmask.
// ---------------------------------------------------------------------------
extern "C" void kernel_launch(void* const* d_in, const int* in_sizes, int n_in,
                              void* d_out, int out_size, void* d_ws, size_t ws_size,
                              hipStream_t stream) {
    (void)n_in; (void)out_size; (void)ws_size;

    const float* edge  = (const float*)d_in[0];
    const float* w     = (const float*)d_in[1];
    const int*   ring  = (const int*)d_in[2];
    const int*   batch = (const int*)d_in[3];
    float*       out   = (float*)d_out;
    unsigned int* mask = (unsigned int*)d_ws;

    const int E = in_sizes[2] / 2;

    // 1) zero 64K mask words (uint4 stores: 16384 threads)
    ring_zero_mask<<<(MASK_WORDS / 4) / 256, 256, 0, stream>>>(mask);
    // 2) scatter edges
    ring_scatter_edges<<<(E + 255) / 256, 256, 0, stream>>>(ring, batch, mask, E);
    // 3) stream 512 MB: CELLS*EMB/4 float4 threads / 256 = 65536 blocks
    ring_stream_add<<<(CELLS * (EMB / 4)) / 256, 256, 0, stream>>>(edge, w, mask, out);
}